// MultiMambaBlockOptimized_21552145891490
// MI455X (gfx1250) — compile-verified
//
#include <hip/hip_runtime.h>
#include <hip/hip_bf16.h>

// ---------------------------------------------------------------------------
// Multi-head Mamba block for gfx1250 (MI455X), fp32 throughout.
// Big GEMMs: LDS-staged double-buffered WMMA (v_wmma_f32_16x16x4_f32) with
// global_load_async_to_lds_b128 staging (ASYNCcnt). Scan runs on VALU with
// 16 fp32 states/channel in registers and Bm/Cm staged in LDS.
// ---------------------------------------------------------------------------

#define B_C   2
#define N_C   2
#define L_C   1024
#define DM_C  256
#define DS_C  16
#define DC_C  4
#define DI_C  512
#define DTR_C 16
#define TOK_C (B_C * N_C * L_C)   // 4096 tokens

typedef float v8f __attribute__((ext_vector_type(8)));
typedef float v2f __attribute__((ext_vector_type(2)));

#define WMMA_F32(a, b, c) \
    __builtin_amdgcn_wmma_f32_16x16x4_f32(false, (a), false, (b), (short)0, (c), false, false)

// Async global->LDS 16-byte copy. INST_OFFSET is added to BOTH the global and
// the LDS address (CDNA5 ISA), used for the +16B B chunk.
__device__ __forceinline__ void async_b128(unsigned lds_off, const float* gaddr) {
    asm volatile("global_load_async_to_lds_b128 %0, %1, off"
                 :: "v"(lds_off), "v"(gaddr) : "memory");
}
__device__ __forceinline__ void async_b128_off16(unsigned lds_off, const float* gaddr) {
    asm volatile("global_load_async_to_lds_b128 %0, %1, off offset:16"
                 :: "v"(lds_off), "v"(gaddr) : "memory");
}
__device__ __forceinline__ void wait_async0() {
    asm volatile("s_wait_asynccnt 0" ::: "memory");
}

// ---------------------------------------------------------------------------
// Tiled WMMA GEMM: C(MxN) = A(MxK) * B(KxN), row-major (lda, ldb=N, ldc=N).
// Block: 256 threads = 8 waves; block tile 64(M) x 128(N); K-step 16.
// Wave (wm,wn) in 2x4 grid computes a 32x32 patch = 2x2 accumulators.
// A staged as [64][20] (pad 20 -> conflict-free ds_load_b64 fragments),
// B staged as [16][128]. Double-buffered async staging.
// Requires: M%64==0, N%128==0, K%16==0.
// ---------------------------------------------------------------------------
#define BM 64
#define BN 128
#define BK 16
#define APAD 20   // A row stride in LDS (floats)

__global__ __launch_bounds__(256)
void wmma_gemm_tiled(const float* __restrict__ A, const float* __restrict__ B,
                     float* __restrict__ C, int M, int N, int K, int lda)
{
    __shared__ float sA[2][BM * APAD];
    __shared__ float sB[2][BK * BN];

    const int tilesN = N / BN;
    const int bx = blockIdx.x % tilesN;
    const int by = blockIdx.x / tilesN;

    const int t = threadIdx.x;

    // --- staging assignments -------------------------------------------------
    // A: 64 rows x 16 floats, 4 contiguous floats / thread (one b128)
    const int arow = t >> 2, ac4 = (t & 3) << 2;
    const float* gA = A + (size_t)(by * BM + arow) * lda + ac4;
    // B: 16 rows x 128 floats, 8 contiguous floats / thread (two b128)
    const int brow = t >> 4, bc8 = (t & 15) << 3;
    const float* gB = B + (size_t)brow * N + bx * BN + bc8;

    unsigned ldsA[2], ldsB[2];
    ldsA[0] = (unsigned)(uintptr_t)&sA[0][arow * APAD + ac4];
    ldsA[1] = (unsigned)(uintptr_t)&sA[1][arow * APAD + ac4];
    ldsB[0] = (unsigned)(uintptr_t)&sB[0][brow * BN + bc8];
    ldsB[1] = (unsigned)(uintptr_t)&sB[1][brow * BN + bc8];

    // --- compute assignments -------------------------------------------------
    const int lane = t & 31;
    const int half = lane >> 4;          // selects K = 2h, 2h+1 within a 4-step
    const int lr   = lane & 15;
    const int wave = t >> 5;
    const int wm   = wave >> 2;          // 0..1  (32-row strip)
    const int wn   = wave & 3;           // 0..3  (32-col strip)

    v8f acc00 = {}, acc01 = {}, acc10 = {}, acc11 = {};

    const int nk = K / BK;

    // prologue: stage k-block 0 into buffer 0
    async_b128(ldsA[0], gA);
    async_b128(ldsB[0], gB);
    async_b128_off16(ldsB[0], gB);
    wait_async0();
    __syncthreads();

    for (int kb = 0; kb < nk; ++kb) {
        const int buf = kb & 1;
        if (kb + 1 < nk) {               // stage next k-block into other buffer
            const float* pa = gA + (kb + 1) * BK;
            const float* pb = gB + (size_t)(kb + 1) * BK * N;
            async_b128(ldsA[buf ^ 1], pa);
            async_b128(ldsB[buf ^ 1], pb);
            async_b128_off16(ldsB[buf ^ 1], pb);
        }

        const float* lA = &sA[buf][0];
        const float* lB = &sB[buf][0];
        const int ra0 = (wm * 32 + lr) * APAD;        // A rows for sub-tile 0
        const int ra1 = ra0 + 16 * APAD;              // ... sub-tile 1
        const int cb0 = wn * 32 + lr;                 // B cols for sub-tile 0

#pragma unroll
        for (int kk = 0; kk < 4; ++kk) {
            const int kp = kk * 4 + 2 * half;
            v2f a0, a1, b0, b1;
            a0.x = lA[ra0 + kp];     a0.y = lA[ra0 + kp + 1];      // ds_load_b64
            a1.x = lA[ra1 + kp];     a1.y = lA[ra1 + kp + 1];
            b0.x = lB[kp * BN + cb0];       b0.y = lB[(kp + 1) * BN + cb0];
            b1.x = lB[kp * BN + cb0 + 16];  b1.y = lB[(kp + 1) * BN + cb0 + 16];
            acc00 = WMMA_F32(a0, b0, acc00);
            acc01 = WMMA_F32(a0, b1, acc01);
            acc10 = WMMA_F32(a1, b0, acc10);
            acc11 = WMMA_F32(a1, b1, acc11);
        }

        wait_async0();
        __syncthreads();
    }

    // --- store: C tile rows = vgpr + 8*half (ISA C layout) -------------------
    const int crow = by * BM + wm * 32 + half * 8;
    const int ccol = bx * BN + wn * 32 + lr;
    float* C0 = C + (size_t)crow * N + ccol;
    float* C1 = C0 + (size_t)16 * N;
#pragma unroll
    for (int v = 0; v < 8; ++v) {
        C0[(size_t)v * N]      = acc00[v];
        C0[(size_t)v * N + 16] = acc01[v];
        C1[(size_t)v * N]      = acc10[v];
        C1[(size_t)v * N + 16] = acc11[v];
    }
}

// ---------------------------------------------------------------------------
// Generic fallback WMMA GEMM (one wave per 16x16 tile) for odd shapes
// (used for the N=48 projection GEMM).
// ---------------------------------------------------------------------------
__global__ void wmma_gemm_f32(const float* __restrict__ A,
                              const float* __restrict__ B,
                              float* __restrict__ C,
                              int M, int N, int K,
                              int lda, int ldb, int ldc)
{
    const int wavesPerBlock = blockDim.x >> 5;
    const int wave   = blockIdx.x * wavesPerBlock + (threadIdx.x >> 5);
    const int tilesN = N >> 4;
    const int tm     = wave / tilesN;
    const int tn     = wave % tilesN;
    if (tm >= (M >> 4)) return;                 // uniform per wave

    const int lane = threadIdx.x & 31;
    const int half = lane >> 4;
    const int lr   = lane & 15;

    const float* Arow = A + (size_t)(tm * 16 + lr) * lda + 2 * half;
    const float* Bcol = B + (size_t)(2 * half) * ldb + tn * 16 + lr;

    v8f acc = {};
    for (int kb = 0; kb < K; kb += 4) {
        v2f a, b;
        a.x = Arow[0];
        a.y = Arow[1];
        b.x = Bcol[0];
        b.y = Bcol[(size_t)ldb];
        acc = WMMA_F32(a, b, acc);
        Arow += 4;
        Bcol += (size_t)4 * ldb;
    }

    float* Crow = C + (size_t)(tm * 16 + half * 8) * ldc + tn * 16 + lr;
#pragma unroll
    for (int v = 0; v < 8; ++v)
        Crow[(size_t)v * ldc] = acc[v];
}

// ---------------------------------------------------------------------------
// Depthwise causal conv (4 taps) + bias + SiLU over the x_in half.
// ---------------------------------------------------------------------------
__global__ void conv_silu_kernel(const float* __restrict__ xr,    // (TOK, 2*DI)
                                 const float* __restrict__ cw,    // (N*DI, 4)
                                 const float* __restrict__ cb,    // (N*DI)
                                 float* __restrict__ out)         // (TOK, DI)
{
    int idx = blockIdx.x * blockDim.x + threadIdx.x;
    if (idx >= TOK_C * DI_C) return;

    const int d  = idx % DI_C;
    const int l  = (idx / DI_C) % L_C;
    const int sq = idx / (DI_C * L_C);
    const int n  = sq % N_C;
    const int ch = n * DI_C + d;

    const float* w = cw + ch * DC_C;
    float acc = cb[ch];
    const size_t tokBase = (size_t)sq * L_C;
#pragma unroll
    for (int j = 0; j < DC_C; ++j) {
        const int ll = l - (DC_C - 1) + j;
        const float v = (ll >= 0) ? xr[(tokBase + ll) * (2 * DI_C) + d] : 0.0f;
        acc = fmaf(v, w[j], acc);
    }
    out[idx] = acc * (1.0f / (1.0f + __expf(-acc)));   // SiLU
}

// ---------------------------------------------------------------------------
// Selective scan fused with softplus(dt)+bias, dA clamp, the reference's
// h[:,0]=dB_u[:,0] override (output only), skip term, and silu(res) gate.
// Grid: one block per sequence (4), 512 threads = one per channel.
// ---------------------------------------------------------------------------
__global__ void scan_kernel(const float* __restrict__ proj,   // (TOK, 48)
                            const float* __restrict__ dtraw,  // (TOK, DI)
                            const float* __restrict__ bdt,    // (DI)
                            const float* __restrict__ Alog,   // (DI, DS)
                            const float* __restrict__ Dp,     // (DI)
                            const float* __restrict__ xconv,  // (TOK, DI)
                            const float* __restrict__ xr,     // (TOK, 2*DI)
                            float* __restrict__ y)            // (TOK, DI)
{
    const int seq = blockIdx.x;
    const int d   = threadIdx.x;

    __shared__ float sB[DS_C];
    __shared__ float sC[DS_C];

    float Arow[DS_C];
#pragma unroll
    for (int s = 0; s < DS_C; ++s)
        Arow[s] = -__expf(Alog[d * DS_C + s]);

    const float Dv = Dp[d];
    const float bd = bdt[d];

    float g[DS_C];
#pragma unroll
    for (int s = 0; s < DS_C; ++s) g[s] = 0.0f;

    for (int t = 0; t < L_C; ++t) {
        const size_t tok = (size_t)seq * L_C + t;

        if (threadIdx.x < 2 * DS_C) {
            const int s = threadIdx.x & (DS_C - 1);
            float* dst = (threadIdx.x < DS_C) ? sB : sC;
            const int off = (threadIdx.x < DS_C) ? DTR_C : (DTR_C + DS_C);
            dst[s] = proj[tok * (DTR_C + 2 * DS_C) + off + s];
        }
        __syncthreads();

        float dtv = dtraw[tok * DI_C + d] + bd;
        dtv = (dtv > 20.0f) ? dtv : __logf(1.0f + __expf(dtv));   // softplus
        const float xv = xconv[tok * DI_C + d];

        float yv = 0.0f;
#pragma unroll
        for (int s = 0; s < DS_C; ++s) {
            float dA  = __expf(dtv * Arow[s]);
            dA        = fmaxf(dA, 1e-10f);
            const float dbu = dtv * sB[s] * xv;
            const float gnew = dA * (g[s] + dbu);
            g[s] = gnew;
            const float h = (t == 0) ? dbu : gnew;
            yv = fmaf(h, sC[s], yv);
        }
        yv = fmaf(xv, Dv, yv);

        const float r  = xr[tok * (2 * DI_C) + DI_C + d];
        const float sr = r * (1.0f / (1.0f + __expf(-r)));        // silu(res)
        y[tok * DI_C + d] = yv * sr;
        __syncthreads();
    }
}

// ---------------------------------------------------------------------------
// Launch
// ---------------------------------------------------------------------------
extern "C" void kernel_launch(void* const* d_in, const int* in_sizes, int n_in,
                              void* d_out, int out_size, void* d_ws, size_t ws_size,
                              hipStream_t stream)
{
    const float* x      = (const float*)d_in[0];
    const float* w_in   = (const float*)d_in[1];
    const float* conv_w = (const float*)d_in[2];
    const float* conv_b = (const float*)d_in[3];
    const float* w_x    = (const float*)d_in[4];
    const float* w_dt   = (const float*)d_in[5];
    const float* b_dt   = (const float*)d_in[6];
    const float* A_log  = (const float*)d_in[7];
    const float* Dp     = (const float*)d_in[8];
    const float* w_out  = (const float*)d_in[9];
    float* out = (float*)d_out;

    float* t_xr   = (float*)d_ws;                                 // TOK x 1024
    float* t_conv = t_xr   + (size_t)TOK_C * (2 * DI_C);          // TOK x 512
    float* t_proj = t_conv + (size_t)TOK_C * DI_C;                // TOK x 48
    float* t_dt   = t_proj + (size_t)TOK_C * (DTR_C + 2 * DS_C);  // TOK x 512
    float* t_y    = t_dt   + (size_t)TOK_C * DI_C;                // TOK x 512

    auto gemm_tiled = [&](const float* A, const float* B, float* C,
                          int M, int N, int K, int lda) {
        const int blocks = (M / BM) * (N / BN);
        hipLaunchKernelGGL(wmma_gemm_tiled, dim3(blocks), dim3(256), 0, stream,
                           A, B, C, M, N, K, lda);
    };

    // 1) x_and_res = x @ w_in          (4096x256)*(256x1024)
    gemm_tiled(x, w_in, t_xr, TOK_C, 2 * DI_C, DM_C, DM_C);

    // 2) depthwise conv + bias + SiLU  -> t_conv
    {
        const int n = TOK_C * DI_C;
        hipLaunchKernelGGL(conv_silu_kernel, dim3((n + 255) / 256), dim3(256), 0, stream,
                           t_xr, conv_w, conv_b, t_conv);
    }

    // 3) proj = x_conv @ w_x           (4096x512)*(512x48)   N=48 -> generic
    {
        const int tiles  = (TOK_C / 16) * ((DTR_C + 2 * DS_C) / 16);
        const int blocks = (tiles + 3) / 4;
        hipLaunchKernelGGL(wmma_gemm_f32, dim3(blocks), dim3(128), 0, stream,
                           t_conv, w_x, t_proj,
                           TOK_C, DTR_C + 2 * DS_C, DI_C,
                           DI_C, DTR_C + 2 * DS_C, DTR_C + 2 * DS_C);
    }

    // 4) dt_raw = proj[:, :16] @ w_dt  (4096x16)*(16x512), lda=48
    gemm_tiled(t_proj, w_dt, t_dt, TOK_C, DI_C, DTR_C, DTR_C + 2 * DS_C);

    // 5) selective scan + gate         -> t_y
    hipLaunchKernelGGL(scan_kernel, dim3(B_C * N_C), dim3(DI_C), 0, stream,
                       t_proj, t_dt, b_dt, A_log, Dp, t_conv, t_xr, t_y);

    // 6) out = y @ w_out               (4096x512)*(512x256)
    gemm_tiled(t_y, w_out, out, TOK_C, DM_C, DI_C, DI_C);
}